// EMLTreeBatched_36996848287825
// MI455X (gfx1250) — compile-verified
//
#include <hip/hip_runtime.h>
#include <math.h>

typedef __attribute__((ext_vector_type(2))) float v2f;
typedef __attribute__((ext_vector_type(4))) float v4f;
typedef __attribute__((ext_vector_type(8))) float v8f;

#define EML_CLAMP 10000.0f

// ---------------------------------------------------------------------------
// Kernel 1: per-s tables. Softmax over leaf_logits (S,32,3) and sigmoid over
// blend_logits (S,31,2). Written both to workspace (for the main kernel) and
// directly into the leaf_probs / gate_probs sections of d_out.
// ---------------------------------------------------------------------------
__global__ __launch_bounds__(64) void eml_setup_kernel(
    const float* __restrict__ leaf_logits,   // (S,32,3)
    const float* __restrict__ blend_logits,  // (S,31,2)
    float* __restrict__ ws_leaf,             // (S,32,3)
    float* __restrict__ ws_gate,             // (S,31,2)
    float* __restrict__ out,
    unsigned long long o_leaf, unsigned long long o_gate)
{
    const int s = blockIdx.x;
    const int t = threadIdx.x;
    if (t < 32) {
        const float* p = leaf_logits + (size_t)(s * 32 + t) * 3;
        float a = p[0], b = p[1], c = p[2];
        float m  = fmaxf(a, fmaxf(b, c));
        float ea = __expf(a - m), eb = __expf(b - m), ec = __expf(c - m);
        float inv = 1.0f / (ea + eb + ec);
        float w0 = ea * inv, w1 = eb * inv, w2 = ec * inv;
        float* wp = ws_leaf + (size_t)(s * 32 + t) * 3;
        wp[0] = w0; wp[1] = w1; wp[2] = w2;
        float* op = out + o_leaf + (size_t)(s * 32 + t) * 3;
        op[0] = w0; op[1] = w1; op[2] = w2;
    } else if (t < 63) {
        const int n = t - 32;  // node 0..30
        const float* p = blend_logits + (size_t)(s * 31 + n) * 2;
        float g0 = 1.0f / (1.0f + __expf(-p[0]));
        float g1 = 1.0f / (1.0f + __expf(-p[1]));
        float* wp = ws_gate + (size_t)(s * 31 + n) * 2;
        wp[0] = g0; wp[1] = g1;
        float* op = out + o_gate + (size_t)(s * 31 + n) * 2;
        op[0] = g0; op[1] = g1;
    }
}

// ---------------------------------------------------------------------------
// eml_exact with gating/bypass: out = exp(log(l) * log(r)) in complex64,
// then nan->0 and clip to +-10000 (matches reference nan_to_num + clip).
// ---------------------------------------------------------------------------
__device__ __forceinline__ void eml_step(float sl, float sr,
                                         float Lre, float Lim,
                                         float Rre, float Rim,
                                         float& ore, float& oim)
{
    const bool bl = sl > 0.99f, br = sr > 0.99f;
    const float oml = 1.0f - sl, omr = 1.0f - sr;
    float lr = bl ? 1.0f : fmaf(oml, Lre, sl);
    float li = bl ? 0.0f : oml * Lim;
    float rr = br ? 1.0f : fmaf(omr, Rre, sr);
    float ri = br ? 0.0f : omr * Rim;
    // complex log: (0.5*log(|z|^2), atan2(im,re))
    float la = 0.5f * __logf(fmaf(lr, lr, li * li));
    float lb = atan2f(li, lr);
    float ra = 0.5f * __logf(fmaf(rr, rr, ri * ri));
    float rb = atan2f(ri, rr);
    // complex multiply of the two logs
    float pr = fmaf(la, ra, -(lb * rb));
    float pi = fmaf(la, rb, lb * ra);
    // complex exp
    float ex = __expf(pr);
    float sn, cs;
    __sincosf(pi, &sn, &cs);
    float cre = ex * cs;
    float cim = ex * sn;
    cre = (cre == cre) ? cre : 0.0f;   // NaN -> 0
    cim = (cim == cim) ? cim : 0.0f;
    ore = fminf(fmaxf(cre, -EML_CLAMP), EML_CLAMP);  // +-inf -> +-C via clip
    oim = fminf(fmaxf(cim, -EML_CLAMP), EML_CLAMP);
}

// Streaming (non-temporal) stores: the complex outputs total ~537MB per call
// (2.8x the 192MB L2) and are never re-read, so keep them out of L2 to
// preserve residency for x/y (re-read S times) and the tables.
// NOTE: __builtin_nontemporal_store requires clang vector types (ext_vector),
// not HIP's struct-based float4/float2.
__device__ __forceinline__ void nt_store4(float* p, float a, float b, float c, float d)
{
    v4f v; v.x = a; v.y = b; v.z = c; v.w = d;
    __builtin_nontemporal_store(v, reinterpret_cast<v4f*>(p));
}
__device__ __forceinline__ void nt_store2(float* p, float a, float b)
{
    v2f v; v.x = a; v.y = b;
    __builtin_nontemporal_store(v, reinterpret_cast<v2f*>(p));
}

// ---------------------------------------------------------------------------
// Kernel 2: main streaming kernel. Block = 256 threads = 8 waves.
// grid = (B/256, S). Each wave computes a 32(batch) x 32(leaf) tile of the
// rank-4 GEMM with 4x V_WMMA_F32_16X16X4_F32, transposes via a wave-private
// padded LDS tile, then each lane evaluates the 5-level eml tree for one
// batch element and emits float4-coalesced non-temporal stores per level.
// ---------------------------------------------------------------------------
__global__ __launch_bounds__(256) void eml_main_kernel(
    const float* __restrict__ xg, const float* __restrict__ yg,
    const float* __restrict__ ws_leaf, const float* __restrict__ ws_gate,
    float* __restrict__ out, int B,
    unsigned long long o_e0, unsigned long long o_e1,
    unsigned long long o_e2, unsigned long long o_e3,
    unsigned long long o_e4)
{
    __shared__ float lw[32 * 3];      // leaf weights for this s
    __shared__ float lg[31 * 2];      // gate sigmoids for this s
    __shared__ float tile[8 * 32 * 33];  // per-wave transpose tiles (stride 33)

    const int s    = blockIdx.y;
    const int tid  = threadIdx.x;
    const int wave = tid >> 5;
    const int lane = tid & 31;
    const int half = lane >> 4;      // 0: lanes 0-15, 1: lanes 16-31
    const int lidx = lane & 15;

    if (tid < 96)              lw[tid]      = ws_leaf[s * 96 + tid];
    else if (tid < 96 + 62)    lg[tid - 96] = ws_gate[s * 62 + (tid - 96)];
    __syncthreads();

    const long long rowBase = (long long)blockIdx.x * 256 + wave * 32;
    float* wtile = &tile[wave * 32 * 33];

    // --- leaf projection via V_WMMA_F32_16X16X4_F32 ---------------------
    #pragma unroll
    for (int nt = 0; nt < 2; ++nt) {
        const int leaf = nt * 16 + lidx;
        // B matrix 4x16 (K x N), K = {w0, w2, w1, 0}
        v2f bm;
        bm.x = half ? lw[leaf * 3 + 1] : lw[leaf * 3 + 0];  // rows K=0 / K=2
        bm.y = half ? 0.0f             : lw[leaf * 3 + 2];  // rows K=1 / K=3
        #pragma unroll
        for (int mt = 0; mt < 2; ++mt) {
            const long long r = rowBase + mt * 16 + lidx;
            const float xv = xg[r];
            const float yv = yg[r];
            // A matrix 16x4 (M x K), row = [1, x, y, 0]
            v2f am;
            am.x = half ? yv   : 1.0f;   // K=0 / K=2
            am.y = half ? 0.0f : xv;     // K=1 / K=3
            v8f c = {};
            v8f d = __builtin_amdgcn_wmma_f32_16x16x4_f32(
                false, am, false, bm, (short)0, c, false, false);
            // D: VGPR r -> M = r + 8*half ; N = lidx
            #pragma unroll
            for (int r8 = 0; r8 < 8; ++r8) {
                const int row = mt * 16 + half * 8 + r8;
                wtile[row * 33 + leaf] = d[r8];
            }
        }
    }
    __syncthreads();

    // --- each lane owns one batch row; run the 5-level eml tree ---------
    const long long b  = rowBase + lane;
    const long long sb = (long long)s * B + b;

    float leaves[32];
    #pragma unroll
    for (int i = 0; i < 32; ++i) leaves[i] = wtile[lane * 33 + i];

    float re[16], im[16];

    // level 0: nodes 0..15, real inputs (imag = 0)
    #pragma unroll
    for (int p = 0; p < 16; ++p)
        eml_step(lg[p * 2 + 0], lg[p * 2 + 1],
                 leaves[2 * p], 0.0f, leaves[2 * p + 1], 0.0f, re[p], im[p]);
    {
        float* e0 = out + o_e0 + sb * 32;
        #pragma unroll
        for (int p = 0; p < 8; ++p)
            nt_store4(e0 + p * 4, re[2 * p], im[2 * p], re[2 * p + 1], im[2 * p + 1]);
    }

    // level 1: nodes 16..23 (in-place: out index p <= input index 2p)
    #pragma unroll
    for (int p = 0; p < 8; ++p)
        eml_step(lg[(16 + p) * 2], lg[(16 + p) * 2 + 1],
                 re[2 * p], im[2 * p], re[2 * p + 1], im[2 * p + 1], re[p], im[p]);
    {
        float* e1 = out + o_e1 + sb * 16;
        #pragma unroll
        for (int p = 0; p < 4; ++p)
            nt_store4(e1 + p * 4, re[2 * p], im[2 * p], re[2 * p + 1], im[2 * p + 1]);
    }

    // level 2: nodes 24..27
    #pragma unroll
    for (int p = 0; p < 4; ++p)
        eml_step(lg[(24 + p) * 2], lg[(24 + p) * 2 + 1],
                 re[2 * p], im[2 * p], re[2 * p + 1], im[2 * p + 1], re[p], im[p]);
    {
        float* e2 = out + o_e2 + sb * 8;
        #pragma unroll
        for (int p = 0; p < 2; ++p)
            nt_store4(e2 + p * 4, re[2 * p], im[2 * p], re[2 * p + 1], im[2 * p + 1]);
    }

    // level 3: nodes 28..29
    #pragma unroll
    for (int p = 0; p < 2; ++p)
        eml_step(lg[(28 + p) * 2], lg[(28 + p) * 2 + 1],
                 re[2 * p], im[2 * p], re[2 * p + 1], im[2 * p + 1], re[p], im[p]);
    nt_store4(out + o_e3 + sb * 4, re[0], im[0], re[1], im[1]);

    // level 4: node 30
    eml_step(lg[30 * 2], lg[30 * 2 + 1], re[0], im[0], re[1], im[1], re[0], im[0]);
    nt_store2(out + o_e4 + sb * 2, re[0], im[0]);
    nt_store2(out + sb * 2,        re[0], im[0]);   // pred section
}

// ---------------------------------------------------------------------------
extern "C" void kernel_launch(void* const* d_in, const int* in_sizes, int n_in,
                              void* d_out, int out_size, void* d_ws, size_t ws_size,
                              hipStream_t stream) {
    const float* x            = (const float*)d_in[0];
    const float* y            = (const float*)d_in[1];
    const float* leaf_logits  = (const float*)d_in[2];
    const float* blend_logits = (const float*)d_in[3];
    float*       out          = (float*)d_out;

    const int B = in_sizes[0];
    const int S = in_sizes[2] / (32 * 3);
    const unsigned long long SB = (unsigned long long)S * (unsigned long long)B;

    // Output layout (floats, complex64 stored interleaved re/im):
    // pred (S,B)c | leaf_probs (S,32,3)f | gate_probs (S,31,2)f |
    // e0 (S,B,16)c | e1 (S,B,8)c | e2 (S,B,4)c | e3 (S,B,2)c | e4 (S,B,1)c
    const unsigned long long o_leaf = 2ULL * SB;
    const unsigned long long o_gate = o_leaf + (unsigned long long)S * 32 * 3;
    const unsigned long long o_e0   = o_gate + (unsigned long long)S * 31 * 2;
    const unsigned long long o_e1   = o_e0 + 2ULL * SB * 16;
    const unsigned long long o_e2   = o_e1 + 2ULL * SB * 8;
    const unsigned long long o_e3   = o_e2 + 2ULL * SB * 4;
    const unsigned long long o_e4   = o_e3 + 2ULL * SB * 2;

    // Workspace: leaf weight table (S*96 floats) then gate table (S*62 floats)
    float* ws_leaf = (float*)d_ws;
    float* ws_gate = ws_leaf + (size_t)S * 96;

    eml_setup_kernel<<<S, 64, 0, stream>>>(leaf_logits, blend_logits,
                                           ws_leaf, ws_gate, out, o_leaf, o_gate);

    dim3 grid(B / 256, S);
    eml_main_kernel<<<grid, 256, 0, stream>>>(x, y, ws_leaf, ws_gate, out, B,
                                              o_e0, o_e1, o_e2, o_e3, o_e4);
}